// VectorQuantizerNG_61718680043736
// MI455X (gfx1250) — compile-verified
//
#include <hip/hip_runtime.h>
#include <math.h>

// ---------------- problem constants ----------------
#define N_PTS   65536
#define EMB_D   128
#define NUM_K   1024
#define TILE_M  16          // rows per workgroup
#define EPSILON 1e-3f
#define CCOST   0.25f

typedef float v2f __attribute__((ext_vector_type(2)));
typedef float v4f __attribute__((ext_vector_type(4)));
typedef float v8f __attribute__((ext_vector_type(8)));

// ---------------- workspace layout (floats) ----------------
#define WS_LOSS   0            // 1 float
#define WS_WNORM  16           // 1024
#define WS_SUMS   (16+1024)    // 1024
#define WS_COUNTS (16+2048)    // 1024
#define WS_HV     (16+3072)    // 1024*128  (16B aligned)
#define WS_TOTAL  (16+3072+NUM_K*EMB_D)   // 134,160 floats

// ---------------- LDS layout (bytes) ----------------
// sDist : 16*1024*4 = 65536
// sIdx  : 16*1024*2 = 32768   (u16 code indices, sorted alongside dist)
// sFlat : 16*128*4  = 8192
// sXn   : 16*4      = 64
// sG    : 1024*4    = 4096    (g[c] = exp(-c/y) lookup table)
// sRed  : 128*4     = 512
#define SMEM_BYTES (65536 + 32768 + 8192 + 64 + 4096 + 512)

// ============================================================
// zero helpers (nontemporal: streaming fills, never re-read soon)
// ============================================================
__global__ void vq_zero2(v2f* p, size_t n2) {
    size_t i = (size_t)blockIdx.x * blockDim.x + threadIdx.x;
    size_t stride = (size_t)gridDim.x * blockDim.x;
    v2f z = {0.f, 0.f};
    for (; i < n2; i += stride) __builtin_nontemporal_store(z, &p[i]);
}

__global__ void vq_zero1(float* p, size_t n) {
    size_t i = (size_t)blockIdx.x * blockDim.x + threadIdx.x;
    size_t stride = (size_t)gridDim.x * blockDim.x;
    for (; i < n; i += stride) p[i] = 0.f;
}

// ============================================================
// wnorm[k] = sum_d weight[k][d]^2
// ============================================================
__global__ void vq_wnorm(const float* __restrict__ weight, float* __restrict__ wnorm) {
    int k = blockIdx.x * blockDim.x + threadIdx.x;
    if (k < NUM_K) {
        const float* w = weight + (size_t)k * EMB_D;
        float s = 0.f;
        for (int d = 0; d < EMB_D; ++d) s = fmaf(w[d], w[d], s);
        wnorm[k] = s;
    }
}

// ============================================================
// main fused kernel: one workgroup = 16 input rows
//   phase 1: WMMA distance strip (16 x 1024) -> LDS (A held in VGPRs)
//   phase 2: stable bitonic sort of each row (dist,idx)
//   phase 3: argmin outputs (one-hot, quantized, loss, counts)
//   phase 4: WMMA h^T @ X tile contributions -> atomicAdd into hv
// ============================================================
__global__ __launch_bounds__(128) void vq_main(
    const float* __restrict__ flat,     // (N, D)
    const float* __restrict__ weight,   // (K, D)
    const int*   __restrict__ timeP,
    const float* __restrict__ wnorm,    // (K,)
    float* __restrict__ outQuant,       // (N, D)
    float* __restrict__ outEnc,         // (N, K) pre-zeroed
    float* __restrict__ wsHv,           // (K, D) accum
    float* __restrict__ wsSums,         // (K,)   accum
    float* __restrict__ wsCounts,       // (K,)   accum
    float* __restrict__ wsLoss)         // (1,)   accum
{
    extern __shared__ char smem_raw[];
    float*          sDist = (float*)smem_raw;                       // 16*1024
    unsigned short* sIdx  = (unsigned short*)(sDist + TILE_M*NUM_K);// 16*1024
    float*          sFlat = (float*)(sIdx + TILE_M*NUM_K);          // 16*128
    float*          sXn   = sFlat + TILE_M*EMB_D;                   // 16
    float*          sG    = sXn + TILE_M;                           // 1024
    float*          sRed  = sG + NUM_K;                             // 128

    const int tid  = threadIdx.x;       // 0..127
    const int lane = tid & 31;
    const int wave = tid >> 5;          // 0..3
    const int row0 = blockIdx.x * TILE_M;

    // neural-gas temperature (time is a device scalar)
    const float tval = (float)(*timeP);
    const float y = 10.0f * powf(1.0e-3f, tval * 0.01f);   // YI*(YF/YI)^(t/100)

    // ---- load 16x128 input tile (16B vectors) ----
    {
        const v4f* src = (const v4f*)(flat + (size_t)row0 * EMB_D);
        v4f* dst = (v4f*)sFlat;
        for (int i = tid; i < TILE_M * EMB_D / 4; i += 128) dst[i] = src[i];
    }
    // ---- g lookup table ----
    for (int c = tid; c < NUM_K; c += 128) sG[c] = expf(-(float)c / y);
    __syncthreads();

    // ---- per-row squared norms ----
    if (tid < TILE_M) {
        float s = 0.f;
        for (int d = 0; d < EMB_D; ++d) {
            float v = sFlat[tid * EMB_D + d];
            s = fmaf(v, v, s);
        }
        sXn[tid] = s;
    }
    __syncthreads();

    // ================= phase 1: distance GEMM via WMMA =================
    // A fragments (this lane's slice of the 16x128 X tile) live in VGPRs
    // and are reused for all 16 code tiles handled by this wave.
    {
        const int mrow  = lane & 15;            // A row / B column within tile
        const int khalf = (lane >> 4) * 2;      // K sub-offset (0 or 2)

        v2f aFrag[EMB_D / 4];
#pragma unroll
        for (int kk = 0; kk < EMB_D / 4; ++kk) {
            const int kb = kk * 4 + khalf;
            aFrag[kk].x = sFlat[mrow * EMB_D + kb];
            aFrag[kk].y = sFlat[mrow * EMB_D + kb + 1];
        }

        for (int tt = 0; tt < 16; ++tt) {
            const int tile  = wave * 16 + tt;   // 0..63
            const int code0 = tile * 16;
            const float* wrow = weight + (size_t)(code0 + mrow) * EMB_D;
            if (tt + 1 < 16) {   // prefetch next weight row (global_prefetch_b8)
                __builtin_prefetch(weight + (size_t)(code0 + 16 + mrow) * EMB_D, 0, 0);
            }
            v8f acc = {};
#pragma unroll
            for (int kk = 0; kk < EMB_D / 4; ++kk) {
                const int kb = kk * 4 + khalf;
                v2f b;
                b.x = wrow[kb];
                b.y = wrow[kb + 1];
                acc = __builtin_amdgcn_wmma_f32_16x16x4_f32(
                        false, aFrag[kk], false, b, (short)0, acc, false, false);
            }
            // scatter distances: C layout -> lane holds (M = i + hi*8, N = lane&15)
            const int nCol = lane & 15;
            const int mAdd = (lane >> 4) * 8;
            const float wn = wnorm[code0 + nCol];
#pragma unroll
            for (int i = 0; i < 8; ++i) {
                const int m = i + mAdd;
                const float dv = sXn[m] + wn - 2.0f * acc[i];
                sDist[m * NUM_K + code0 + nCol] = dv;
                sIdx [m * NUM_K + code0 + nCol] = (unsigned short)(code0 + nCol);
            }
        }
    }

    // ================= phase 2: stable bitonic sort per row =================
    // 16 rows x 512 pairs = 8192 pairs per step, 128 threads
    for (unsigned k = 2; k <= NUM_K; k <<= 1) {
        for (unsigned j = k >> 1; j > 0; j >>= 1) {
            __syncthreads();
            for (int p = tid; p < TILE_M * (NUM_K / 2); p += 128) {
                const int row = p >> 9;
                const unsigned pr = (unsigned)(p & 511);
                const unsigned i   = pr + (pr & ~(j - 1u));
                const unsigned ixj = i ^ j;
                float* dr = sDist + row * NUM_K;
                unsigned short* xr = sIdx + row * NUM_K;
                const float d0 = dr[i],  d1 = dr[ixj];
                const unsigned short x0 = xr[i], x1 = xr[ixj];
                const bool up = ((i & k) == 0);
                const bool gt = (d0 > d1) || ((d0 == d1) && (x0 > x1));
                if (gt == up) {   // swap (lexicographic, indices unique)
                    dr[i] = d1; dr[ixj] = d0;
                    xr[i] = x1; xr[ixj] = x0;
                }
            }
        }
    }
    __syncthreads();

    // ================= phase 3: argmin outputs =================
    {
        const int row = tid >> 3;        // 0..15
        const int sub = tid & 7;         // 0..7
        const int amin = (int)sIdx[row * NUM_K + 0];  // rank 0 == stable argmin
        const float* wq = weight + (size_t)amin * EMB_D;
        float lossPart = 0.f;
        for (int d = sub; d < EMB_D; d += 8) {
            const float f = sFlat[row * EMB_D + d];
            const float q = wq[d];
            const float diff = q - f;
            lossPart = fmaf(diff, diff, lossPart);
            // straight-through estimator: flat + (quantized - flat); streamed out
            __builtin_nontemporal_store(f + (q - f),
                &outQuant[(size_t)(row0 + row) * EMB_D + d]);
        }
        sRed[tid] = lossPart;
        if (sub == 0) {
            __builtin_nontemporal_store(1.0f,
                &outEnc[(size_t)(row0 + row) * NUM_K + amin]);
            atomicAdd(&wsCounts[amin], 1.0f);
        }
    }
    __syncthreads();
    if (tid == 0) {
        float s = 0.f;
        for (int i = 0; i < 128; ++i) s += sRed[i];
        atomicAdd(wsLoss, s);
    }

    // ---- rank sums: sums[r] += sum_n g(ord[n][r]) ----
    for (int r = tid; r < NUM_K; r += 128) {
        float s = 0.f;
        for (int n = 0; n < TILE_M; ++n) s += sG[sIdx[n * NUM_K + r]];
        atomicAdd(&wsSums[r], s);
    }

    // ================= phase 4: hv += h^T @ X via WMMA =================
    // hv[r][d] += sum_n g(ord[n][r]) * x[n][d]
    // d-tile outer (2 per wave): B fragments (flat) hoisted to VGPRs,
    // reused across all 64 rank tiles.
    {
        const int m     = lane & 15;        // rank within tile (A), d within tile (B)
        const int khalf = (lane >> 4) * 2;
        for (int dt = 0; dt < 2; ++dt) {
            const int d0 = (wave * 2 + dt) * 16;
            v2f bFrag[TILE_M / 4];
#pragma unroll
            for (int kk = 0; kk < TILE_M / 4; ++kk) {
                const int n0 = kk * 4 + khalf;
                bFrag[kk].x = sFlat[(n0 + 0) * EMB_D + d0 + m];
                bFrag[kk].y = sFlat[(n0 + 1) * EMB_D + d0 + m];
            }
            for (int rt = 0; rt < 64; ++rt) {
                const int rank0 = rt * 16;
                v8f acc = {};
#pragma unroll
                for (int kk = 0; kk < TILE_M / 4; ++kk) {
                    const int n0 = kk * 4 + khalf;  // row index (K dim of WMMA)
                    v2f a;
                    a.x = sG[sIdx[(n0 + 0) * NUM_K + rank0 + m]];
                    a.y = sG[sIdx[(n0 + 1) * NUM_K + rank0 + m]];
                    acc = __builtin_amdgcn_wmma_f32_16x16x4_f32(
                            false, a, false, bFrag[kk], (short)0, acc, false, false);
                }
                const int nC   = lane & 15;
                const int mAdd = (lane >> 4) * 8;
#pragma unroll
                for (int i = 0; i < 8; ++i) {
                    const int rk = rank0 + i + mAdd;
                    atomicAdd(&wsHv[(size_t)rk * EMB_D + d0 + nC], acc[i]);
                }
            }
        }
    }
}

// ============================================================
// update / new_weight
// ============================================================
__global__ void vq_finalize_update(
    const float* __restrict__ weight,
    const float* __restrict__ hv,
    const float* __restrict__ sums,
    float* __restrict__ outNewW,
    float* __restrict__ outUpd)
{
    const int k = blockIdx.x;          // 0..1023
    const int d = threadIdx.x;         // 0..127
    const size_t off = (size_t)k * EMB_D + d;
    const float w = weight[off];
    const float u = EPSILON * (hv[off] - sums[k] * w);
    __builtin_nontemporal_store(u,     &outUpd[off]);
    __builtin_nontemporal_store(w + u, &outNewW[off]);
}

// ============================================================
// loss + perplexity
// ============================================================
__global__ void vq_finalize_scalars(
    const float* __restrict__ counts,
    const float* __restrict__ lossAcc,
    float* __restrict__ outLoss,
    float* __restrict__ outPerp)
{
    __shared__ float red[NUM_K];
    const int t = threadIdx.x;         // 0..1023
    const float p = counts[t] * (1.0f / (float)N_PTS);
    red[t] = p * logf(p + 1e-10f);
    __syncthreads();
    for (int s = NUM_K / 2; s > 0; s >>= 1) {
        if (t < s) red[t] += red[t + s];
        __syncthreads();
    }
    if (t == 0) {
        *outPerp = expf(-red[0]);
        *outLoss = CCOST * (*lossAcc) / ((float)N_PTS * (float)EMB_D);
    }
}

// ============================================================
extern "C" void kernel_launch(void* const* d_in, const int* in_sizes, int n_in,
                              void* d_out, int out_size, void* d_ws, size_t ws_size,
                              hipStream_t stream) {
    const float* flat   = (const float*)d_in[0];   // (65536,128) f32
    const float* weight = (const float*)d_in[1];   // (1024,128) f32
    const int*   timeP  = (const int*)d_in[2];     // scalar int

    // d_out layout (reference tuple order, flattened):
    float* out      = (float*)d_out;
    float* outLoss  = out;                                     // 1
    float* outQuant = out + 1;                                 // N*D
    float* outPerp  = out + 1 + (size_t)N_PTS * EMB_D;         // 1
    float* outEnc   = out + 2 + (size_t)N_PTS * EMB_D;         // N*K
    float* outNewW  = outEnc + (size_t)N_PTS * NUM_K;          // K*D
    float* outUpd   = outNewW + (size_t)NUM_K * EMB_D;         // K*D

    float* ws       = (float*)d_ws;
    float* wsLoss   = ws + WS_LOSS;
    float* wsWnorm  = ws + WS_WNORM;
    float* wsSums   = ws + WS_SUMS;
    float* wsCounts = ws + WS_COUNTS;
    float* wsHv     = ws + WS_HV;

    // 1) zero accumulators + one-hot encodings region
    vq_zero1<<<256, 256, 0, stream>>>(ws, (size_t)WS_TOTAL);
    vq_zero2<<<8192, 256, 0, stream>>>((v2f*)outEnc,
                                       (size_t)N_PTS * NUM_K / 2);  // 8B aligned

    // 2) codebook norms
    vq_wnorm<<<NUM_K / 128, 128, 0, stream>>>(weight, wsWnorm);

    // 3) fused distance-GEMM / sort / outputs / neural-gas accumulation
    vq_main<<<N_PTS / TILE_M, 128, SMEM_BYTES, stream>>>(
        flat, weight, timeP, wsWnorm,
        outQuant, outEnc, wsHv, wsSums, wsCounts, wsLoss);

    // 4) weight update outputs
    vq_finalize_update<<<NUM_K, EMB_D, 0, stream>>>(
        weight, wsHv, wsSums, outNewW, outUpd);

    // 5) scalar outputs
    vq_finalize_scalars<<<1, NUM_K, 0, stream>>>(
        wsCounts, wsLoss, outLoss, outPerp);
}